// ECELoss_DRL_75462575391108
// MI455X (gfx1250) — compile-verified
//
#include <hip/hip_runtime.h>
#include <stdint.h>

#define NBINS 20
#define WPB 8              // waves per block
#define TPB (WPB * 32)
#define RPB 6              // rows staged per wave per batch (6*512B = 3KB per buffer)
#define NCOL 128           // C (columns), fixed by the reference setup

// d_ws layout (as u64[]):
//   ws[0..19]   : (count << 32) | acc_sum   per bin
//   ws[20..39]  : conf_sum in 2^40 fixed point per bin
__global__ void ece_zero_kernel(unsigned long long* ws) {
    int t = threadIdx.x;
    if (t < 2 * NBINS) ws[t] = 0ull;
}

__launch_bounds__(TPB)
__global__ void ece_main_kernel(const float* __restrict__ sm,
                                const int* __restrict__ labels,
                                unsigned long long* __restrict__ ws,
                                int N) {
    // Staging: WPB waves * 2 buffers * RPB rows * NCOL floats = 48KB
    __shared__ float stage[WPB * 2 * RPB * NCOL];
    __shared__ unsigned int s_cnt[NBINS];
    __shared__ unsigned int s_acc[NBINS];
    __shared__ unsigned long long s_conf[NBINS];

    const int tid  = threadIdx.x;
    const int lane = tid & 31;
    const int wave = tid >> 5;

    if (tid < NBINS) { s_cnt[tid] = 0u; s_acc[tid] = 0u; s_conf[tid] = 0ull; }
    __syncthreads();

    const int totalWaves = gridDim.x * WPB;
    const int waveId     = blockIdx.x * WPB + wave;
    const int numBatches = (N + RPB - 1) / RPB;

    // LDS byte offset of this wave's staging area (generic ptr low 32 bits = LDS offset)
    const uint32_t ldsWaveBase = (uint32_t)(size_t)(&stage[wave * (2 * RPB * NCOL)]);
    const uint32_t laneByte    = (uint32_t)lane * 16u;   // 16B (float4) per lane per row

    int b   = waveId;
    int cur = 0;
    if (b < numBatches) {
        // ---- prologue: async-stage batch b into buffer 0 ----
        {
#pragma unroll
            for (int r = 0; r < RPB; ++r) {
                int row = b * RPB + r; if (row >= N) row = N - 1;   // clamp (safe dup)
                uint32_t voff = (uint32_t)row * (NCOL * 4) + laneByte;
                uint32_t dst  = ldsWaveBase + (uint32_t)(r * NCOL * 4) + laneByte;
                asm volatile("global_load_async_to_lds_b128 %0, %1, %2"
                             :: "v"(dst), "v"(voff), "s"(sm) : "memory");
            }
        }
        while (b < numBatches) {
            const int nb = b + totalWaves;

            // coalesced label fetch for this batch (lane r -> row b*RPB+r)
            int myLbl = -1;
            {
                int lrow = b * RPB + lane;
                if (lane < RPB && lrow < N) myLbl = labels[lrow];
            }

            if (nb < numBatches) {
                // ---- async-stage next batch into the other buffer ----
                const uint32_t nBase = ldsWaveBase + (uint32_t)((cur ^ 1) * RPB * NCOL * 4);
#pragma unroll
                for (int r = 0; r < RPB; ++r) {
                    int row = nb * RPB + r; if (row >= N) row = N - 1;
                    uint32_t voff = (uint32_t)row * (NCOL * 4) + laneByte;
                    uint32_t dst  = nBase + (uint32_t)(r * NCOL * 4) + laneByte;
                    asm volatile("global_load_async_to_lds_b128 %0, %1, %2"
                                 :: "v"(dst), "v"(voff), "s"(sm) : "memory");
                }
                // 12 outstanding; in-order completion => <=6 means current batch is ready
                asm volatile("s_wait_asynccnt 0x6" ::: "memory");
            } else {
                asm volatile("s_wait_asynccnt 0x0" ::: "memory");
            }

            // ---- consume current batch from LDS ----
            const float4* buf =
                (const float4*)&stage[wave * (2 * RPB * NCOL) + cur * RPB * NCOL];

            float        myConf  = 0.0f;
            int          myBin   = 0;
            unsigned int myAcc   = 0u;
            int          myValid = 0;

#pragma unroll
            for (int r = 0; r < RPB; ++r) {
                float4 v = buf[r * (NCOL / 4) + lane];          // ds_load_b128
                // wave max (value) via butterfly
                float wm = fmaxf(fmaxf(v.x, v.y), fmaxf(v.z, v.w));
#pragma unroll
                for (int off = 16; off >= 1; off >>= 1)
                    wm = fmaxf(wm, __shfl_xor(wm, off, 32));
                // first-occurrence argmax: per-lane lowest matching slot...
                int base4 = lane * 4;
                int cand  = 0x7fffffff;
                if (v.w == wm) cand = base4 + 3;
                if (v.z == wm) cand = base4 + 2;
                if (v.y == wm) cand = base4 + 1;
                if (v.x == wm) cand = base4 + 0;
                // ...then the cand of the lowest lane holding any match
                unsigned long long m = __ballot(cand != 0x7fffffff);
                int src = __ffsll(m) - 1;        // lowest matching lane (m != 0 always)
                int wi  = __shfl(cand, src);     // global argmax, first occurrence

                const int row = b * RPB + r;
                if (lane == r) {                 // lane r carries row r's result
                    myValid = (row < N);
                    myConf  = wm;
                    myAcc   = (wi == myLbl) ? 1u : 0u;
                    int bb  = (int)ceilf(wm * 20.0f) - 1;   // matches jnp.ceil path
                    myBin   = min(max(bb, 0), NBINS - 1);
                }
            }
            // one atomic round covers all RPB rows (lanes 0..RPB-1 active)
            if (lane < RPB && myValid) {
                // exact 2^40 fixed point (conf >= 1/128, so conversion is exact)
                unsigned long long cf =
                    (unsigned long long)(myConf * 1099511627776.0f);
                atomicAdd(&s_cnt[myBin], 1u);
                atomicAdd(&s_acc[myBin], myAcc);
                atomicAdd(&s_conf[myBin], cf);
            }
            b = nb; cur ^= 1;
        }
    }
    __syncthreads();

    // ---- flush block bins to global scratch ----
    if (tid < NBINS) {
        unsigned long long ca =
            ((unsigned long long)s_cnt[tid] << 32) | (unsigned long long)s_acc[tid];
        if (ca) atomicAdd(&ws[tid], ca);
        unsigned long long cs = s_conf[tid];
        if (cs) atomicAdd(&ws[NBINS + tid], cs);
    }
}

__global__ void ece_final_kernel(const unsigned long long* __restrict__ ws,
                                 float* __restrict__ out, int N) {
    int t = threadIdx.x;
    if (t < NBINS) {
        unsigned long long ca = ws[t];
        unsigned int cnt = (unsigned int)(ca >> 32);
        unsigned int acc = (unsigned int)(ca & 0xffffffffu);
        out[1 + t] = cnt ? (float)((double)acc / (double)cnt) : 0.0f;
    }
    if (t == 0) {
        double ece = 0.0;
        for (int i = 0; i < NBINS; ++i) {
            unsigned long long ca = ws[i];
            unsigned int cnt = (unsigned int)(ca >> 32);
            if (cnt) {
                unsigned int acc = (unsigned int)(ca & 0xffffffffu);
                double conf_sum = (double)ws[NBINS + i] / 1099511627776.0;
                double avg_conf = conf_sum / (double)cnt;
                double avg_acc  = (double)acc / (double)cnt;
                double prop     = (double)cnt / (double)N;
                ece += fabs(avg_conf - avg_acc) * prop;
            }
        }
        out[0] = (float)ece;
    }
}

extern "C" void kernel_launch(void* const* d_in, const int* in_sizes, int n_in,
                              void* d_out, int out_size, void* d_ws, size_t ws_size,
                              hipStream_t stream) {
    const float* sm     = (const float*)d_in[0];   // softmaxes [N, 128] f32
    const int*   labels = (const int*)d_in[1];     // labels [N] i32
    const int    N      = in_sizes[1];
    unsigned long long* ws = (unsigned long long*)d_ws;
    float* out = (float*)d_out;                    // [0]=ece, [1..20]=ys

    ece_zero_kernel<<<1, 64, 0, stream>>>(ws);
    ece_main_kernel<<<2048, TPB, 0, stream>>>(sm, labels, ws, N);
    ece_final_kernel<<<1, 32, 0, stream>>>(ws, out, N);
}